// DotHash_35175782154944
// MI455X (gfx1250) — compile-verified
//
#include <hip/hip_runtime.h>

// ---------------------------------------------------------------------------
// DotHash motif counting for MI455X (gfx1250, wave32).
//
// Pipeline:
//   ws: [ one_hop (N*256 f32) | two_hop (N*256 f32) | deg (N f32) ]
//   K1: zero ws (hipMemsetAsync)
//   K2: scatter: one_hop[row] += node_vectors[col]*w[col]; deg[row] += 1
//   K3: scatter: two_hop[row] += one_hop[col]
//   K4: WMMA (v_wmma_f32_16x16x4_f32) batched dot products, 16 edges/wave.
// ---------------------------------------------------------------------------

#define DIMK 256

typedef __attribute__((ext_vector_type(2))) float v2f;
typedef __attribute__((ext_vector_type(8))) float v8f;

// ---------------------------------------------------------------------------
// Scatter-add SpMM: one wave (32 lanes) per edge, 8 dims per lane.
// weight==nullptr -> scale 1.0 ; deg==nullptr -> skip degree accumulation.
// ---------------------------------------------------------------------------
__global__ void dh_scatter_kernel(const int* __restrict__ rows,
                                  const int* __restrict__ cols,
                                  const float* __restrict__ src,
                                  const float* __restrict__ weight,
                                  float* __restrict__ dst,
                                  float* __restrict__ deg,
                                  int nE) {
    long long tid = (long long)blockIdx.x * blockDim.x + threadIdx.x;
    int e    = (int)(tid >> 5);
    int lane = (int)(tid & 31);
    if (e >= nE) return;

    int row = rows[e];
    int col = cols[e];
    float w = (weight != nullptr) ? weight[col] : 1.0f;

    const float* sp = src + (size_t)col * DIMK + lane * 8;
    float*       dp = dst + (size_t)row * DIMK + lane * 8;

    float4 v0 = *(const float4*)(sp);
    float4 v1 = *(const float4*)(sp + 4);

    atomicAdd(dp + 0, v0.x * w);
    atomicAdd(dp + 1, v0.y * w);
    atomicAdd(dp + 2, v0.z * w);
    atomicAdd(dp + 3, v0.w * w);
    atomicAdd(dp + 4, v1.x * w);
    atomicAdd(dp + 5, v1.y * w);
    atomicAdd(dp + 6, v1.z * w);
    atomicAdd(dp + 7, v1.w * w);

    if (deg != nullptr && lane == 0) {
        atomicAdd(deg + row, 1.0f);
    }
}

// Extract element r (0..7) of a WMMA f32 accumulator fragment.
__device__ __forceinline__ float dh_extract(v8f v, int r) {
    switch (r) {
        case 0: return v[0];
        case 1: return v[1];
        case 2: return v[2];
        case 3: return v[3];
        case 4: return v[4];
        case 5: return v[5];
        case 6: return v[6];
        default: return v[7];
    }
}

__device__ __forceinline__ v8f dh_wmma(v2f a, v2f b, v8f c) {
    // V_WMMA_F32_16X16X4_F32: D = A(16x4) x B(4x16) + C(16x16), all f32.
    return __builtin_amdgcn_wmma_f32_16x16x4_f32(
        /*neg_a=*/false, a, /*neg_b=*/false, b,
        /*c_mod=*/(short)0, c, /*reuse_a=*/false, /*reuse_b=*/false);
}

// ---------------------------------------------------------------------------
// Batched dot products via f32 WMMA. One wave handles 16 query edges.
//
// A-fragment layout (ISA 7.12.2, 32-bit A 16x4): lane m (m = lane&15) holds
// row M=m, K = k0 + (lane>>4)*2 .. +1  -> one float2 load per operand.
// B-fragment assumed lane-symmetric (lane n holds col N=n, same K split),
// so a register loaded from row r serves as both the A-role and B-role
// fragment of r (used for the self dots OS.TS and OT.TT).
//
// C/D diag(i): i<8 -> VGPR i, lane i ; i>=8 -> VGPR i-8, lane i+16.
// ---------------------------------------------------------------------------
__global__ void dh_dot_wmma_kernel(const int* __restrict__ edges,      // [2*EQ]
                                   const float* __restrict__ one_hop,
                                   const float* __restrict__ two_hop,
                                   const float* __restrict__ node_vectors,
                                   const float* __restrict__ node_weight,
                                   const float* __restrict__ deg,
                                   float* __restrict__ out,            // [4*EQ]
                                   int eq) {
    int wave = (int)((blockIdx.x * (long long)blockDim.x + threadIdx.x) >> 5);
    int lane = threadIdx.x & 31;
    int base = wave * 16;
    if (base >= eq) return;   // wave-uniform: EXEC stays all-ones for WMMA

    int m     = lane & 15;         // row/col index within the 16-edge batch
    int khalf = (lane >> 4) * 2;   // K sub-offset inside each 4-wide chunk

    int e = base + m;
    int s = edges[e];
    int t = edges[eq + e];

    const float* os = one_hop      + (size_t)s * DIMK;
    const float* ts = two_hop      + (size_t)s * DIMK;
    const float* xs = node_vectors + (size_t)s * DIMK;
    const float* ot = one_hop      + (size_t)t * DIMK;
    const float* tt = two_hop      + (size_t)t * DIMK;
    const float* xt = node_vectors + (size_t)t * DIMK;

    // a_adj = two_hop[s] - deg[s]*w[s]*node_vectors[s]  (since x = nv*w)
    float cs = deg[s] * node_weight[s];
    float ct = deg[t] * node_weight[t];

    v8f c11  = {};   // OS . OT
    v8f c12a = {};   // OS . TT
    v8f c12b = {};   // TS . OT
    v8f c22  = {};   // (TS - cs*XS) . (TT - ct*XT)
    v8f cse1 = {};   // OS . TS
    v8f cse2 = {};   // OT . TT

    for (int k0 = 0; k0 < DIMK; k0 += 4) {
        int k = k0 + khalf;
        v2f a_os = *(const v2f*)(os + k);
        v2f a_ts = *(const v2f*)(ts + k);
        v2f a_xs = *(const v2f*)(xs + k);
        v2f b_ot = *(const v2f*)(ot + k);
        v2f b_tt = *(const v2f*)(tt + k);
        v2f b_xt = *(const v2f*)(xt + k);

        v2f a_adj, b_adj;
        a_adj[0] = a_ts[0] - cs * a_xs[0];
        a_adj[1] = a_ts[1] - cs * a_xs[1];
        b_adj[0] = b_tt[0] - ct * b_xt[0];
        b_adj[1] = b_tt[1] - ct * b_xt[1];

        c11  = dh_wmma(a_os,  b_ot, c11);
        c12a = dh_wmma(a_os,  b_tt, c12a);
        c12b = dh_wmma(a_ts,  b_ot, c12b);
        c22  = dh_wmma(a_adj, b_adj, c22);
        cse1 = dh_wmma(a_os,  a_ts, cse1);   // B-role reuse of the s rows
        cse2 = dh_wmma(b_ot,  b_tt, cse2);   // B-role reuse of the t rows
    }

    // Pull the diagonals: lanes 0..7 own edges 0..7, lanes 24..31 own 8..15.
    int i = -1, r = 0;
    if (lane < 8)        { i = lane;      r = lane;      }
    else if (lane >= 24) { i = lane - 16; r = lane - 24; }

    if (i >= 0) {
        int eo = base + i;
        if (eo < eq) {
            out[eo]          = dh_extract(c11, r);                        // count_1_1
            out[eq + eo]     = dh_extract(c12a, r) + dh_extract(c12b, r); // count_1_2
            out[2 * eq + eo] = dh_extract(c22, r);                        // count_2_2
            out[3 * eq + eo] = dh_extract(cse1, r) + dh_extract(cse2, r); // count_self_1_2
        }
    }
}

// ---------------------------------------------------------------------------
// Host launcher
// ---------------------------------------------------------------------------
extern "C" void kernel_launch(void* const* d_in, const int* in_sizes, int n_in,
                              void* d_out, int out_size, void* d_ws, size_t ws_size,
                              hipStream_t stream) {
    const int*   edges        = (const int*)d_in[0];    // [2, EQ] int32
    const int*   adj_row      = (const int*)d_in[1];    // [E] int32
    const int*   adj_col      = (const int*)d_in[2];    // [E] int32
    const float* node_weight  = (const float*)d_in[3];  // [N] f32
    const float* node_vectors = (const float*)d_in[4];  // [N, 256] f32

    int eq = in_sizes[0] / 2;
    int e  = in_sizes[1];
    int n  = in_sizes[3];

    float* one_hop = (float*)d_ws;
    float* two_hop = one_hop + (size_t)n * DIMK;
    float* deg     = two_hop + (size_t)n * DIMK;

    // Zero one_hop + two_hop + deg (atomics accumulate into them).
    size_t zero_bytes = ((size_t)n * DIMK * 2 + (size_t)n) * sizeof(float);
    hipMemsetAsync(d_ws, 0, zero_bytes, stream);

    // Scatter hops: one wave per edge, 32 lanes x 8 dims.
    long long sc_threads = (long long)e * 32;
    dim3 blk(256);
    dim3 grd_sc((unsigned)((sc_threads + 255) / 256));
    dh_scatter_kernel<<<grd_sc, blk, 0, stream>>>(
        adj_row, adj_col, node_vectors, node_weight, one_hop, deg, e);
    dh_scatter_kernel<<<grd_sc, blk, 0, stream>>>(
        adj_row, adj_col, one_hop, nullptr, two_hop, nullptr, e);

    // WMMA dot stage: one wave per 16 query edges, 4 waves per block.
    int waves = (eq + 15) / 16;
    dim3 blk2(128);
    dim3 grd2((unsigned)(((long long)waves * 32 + 127) / 128));
    dh_dot_wmma_kernel<<<grd2, blk2, 0, stream>>>(
        edges, one_hop, two_hop, node_vectors, node_weight, deg,
        (float*)d_out, eq);
}